// CoupledFEMSolver_58909771432903
// MI455X (gfx1250) — compile-verified
//
#include <hip/hip_runtime.h>
#include <hip/hip_bf16.h>

// ---------------- problem constants (match reference) ----------------
#define N_NODES_C   8000
#define N_SOLID_C   2000
#define NF_ELEM_C   40000
#define NS_ELEM_C   10000
#define N_IFACE_C   400
#define N_NEAR_C    200
#define NT_C        (N_NODES_C + 3 * N_SOLID_C)   /* 14000 */
#define NB_C        N_NODES_C                      /* solid block offset */

static constexpr double PI_D      = 3.14159265358979323846;
static constexpr double OMEGA_D   = 2.0 * PI_D * 1000.0;
static constexpr float  K2F       = (float)((OMEGA_D / 343.0) * (OMEGA_D / 343.0)); // (omega/c_f)^2
static constexpr float  OM2F      = (float)(OMEGA_D * OMEGA_D);
static constexpr float  PENALTY_F = 1.0e8f;
static constexpr float  P0_F      = 1.0f;

typedef __attribute__((ext_vector_type(2))) float v2f;
typedef __attribute__((ext_vector_type(8))) float v8f;

// ---------------------------------------------------------------------
// Kernel 1: zero the whole output (A_g 14000x14000 + F_g 14000).
// This is the bandwidth-dominant step (~784 MB of stores).
// ---------------------------------------------------------------------
__global__ void fem_zero_out(float4* __restrict__ out, long long n4) {
    long long i      = (long long)blockIdx.x * blockDim.x + threadIdx.x;
    long long stride = (long long)gridDim.x * blockDim.x;
    float4 z; z.x = 0.f; z.y = 0.f; z.z = 0.f; z.w = 0.f;
    for (; i < n4; i += stride) out[i] = z;
}

// ---------------------------------------------------------------------
// Kernel 2: fluid tet assembly. One thread per element, 16 atomic adds:
//   Ke[i][j] = V * (g_i . g_j),  Me[i][j] = V/10 * (i==j ? 3 : 1)
//   contribution = Ke - k^2 * Me
// ---------------------------------------------------------------------
__global__ void fem_fluid_assemble(const float* __restrict__ nodes,
                                   const int*   __restrict__ elems,
                                   float*       __restrict__ A) {
    int e = blockIdx.x * blockDim.x + threadIdx.x;
    if (e >= NF_ELEM_C) return;

    int id[4];
#pragma unroll
    for (int i = 0; i < 4; ++i) id[i] = elems[e * 4 + i];

    float X[4], Y[4], Z[4];
#pragma unroll
    for (int i = 0; i < 4; ++i) {
        const float* p = nodes + (size_t)id[i] * 3;
        X[i] = p[0]; Y[i] = p[1]; Z[i] = p[2];
    }
    float e1x = X[1]-X[0], e1y = Y[1]-Y[0], e1z = Z[1]-Z[0];
    float e2x = X[2]-X[0], e2y = Y[2]-Y[0], e2z = Z[2]-Z[0];
    float e3x = X[3]-X[0], e3y = Y[3]-Y[0], e3z = Z[3]-Z[0];

    // grad N1 = (e2 x e3)/det, N2 = (e3 x e1)/det, N3 = (e1 x e2)/det, N0 = -(sum)
    float c1x = e2y*e3z - e2z*e3y, c1y = e2z*e3x - e2x*e3z, c1z = e2x*e3y - e2y*e3x;
    float det = e1x*c1x + e1y*c1y + e1z*c1z;
    float inv = 1.0f / det;
    float c2x = e3y*e1z - e3z*e1y, c2y = e3z*e1x - e3x*e1z, c2z = e3x*e1y - e3y*e1x;
    float c3x = e1y*e2z - e1z*e2y, c3y = e1z*e2x - e1x*e2z, c3z = e1x*e2y - e1y*e2x;

    float gx[4], gy[4], gz[4];
    gx[1] = c1x*inv; gy[1] = c1y*inv; gz[1] = c1z*inv;
    gx[2] = c2x*inv; gy[2] = c2y*inv; gz[2] = c2z*inv;
    gx[3] = c3x*inv; gy[3] = c3y*inv; gz[3] = c3z*inv;
    gx[0] = -(gx[1]+gx[2]+gx[3]);
    gy[0] = -(gy[1]+gy[2]+gy[3]);
    gz[0] = -(gz[1]+gz[2]+gz[3]);

    float V    = fabsf(det) * (1.0f / 6.0f);
    float mfac = K2F * V * 0.1f;   // k^2 * V/10

#pragma unroll
    for (int i = 0; i < 4; ++i) {
        size_t rowb = (size_t)id[i] * NT_C;
#pragma unroll
        for (int j = 0; j < 4; ++j) {
            float val = V * (gx[i]*gx[j] + gy[i]*gy[j] + gz[i]*gz[j])
                      - mfac * ((i == j) ? 3.0f : 1.0f);
            atomicAdd(&A[rowb + id[j]], val);
        }
    }
}

// ---------------------------------------------------------------------
// Kernel 3: Dirichlet rows (after fluid assembly): zero fluid columns of
// each near row, set diag = 1, set F = P0.
// ---------------------------------------------------------------------
__global__ void fem_dirichlet(const int* __restrict__ near_idx,
                              float* __restrict__ A,
                              float* __restrict__ F) {
    int row = near_idx[blockIdx.y];
    int col = blockIdx.x * blockDim.x + threadIdx.x;
    if (col < N_NODES_C)
        A[(size_t)row * NT_C + col] = (col == row) ? 1.0f : 0.0f;
    if (blockIdx.x == 0 && threadIdx.x == 0)
        F[row] = P0_F;
}

// ---------------------------------------------------------------------
// Kernel 4: fluid coupling penalty on diagonal (duplicates accumulate).
// ---------------------------------------------------------------------
__global__ void fem_fluid_penalty(const int* __restrict__ imap,
                                  float* __restrict__ A) {
    int i = blockIdx.x * blockDim.x + threadIdx.x;
    if (i >= N_IFACE_C) return;
    int nd = imap[i];
    atomicAdd(&A[(size_t)nd * NT_C + nd], PENALTY_F);
}

// ---------------------------------------------------------------------
// Kernel 5: solid tet assembly via WMMA. One wave32 per element.
//   KeS = Vs * B^T (D B)  done as two v_wmma_f32_16x16x4_f32 ops on
//   zero-padded operands: A-op = B^T (12x6 -> 16x8), B-op = D*B (6x12 -> 8x16).
//   Lumped mass subtracted on local diagonal; 12x12 scattered with atomics.
// ISA fragment layouts used:
//   A 16x4 f32: lanes 0-15 hold M=lane, VGPR0=K0,VGPR1=K1; lanes 16-31 K2,K3.
//   B 4x16 f32: lanes 0-15 hold N=lane, VGPR0=K0,VGPR1=K1; lanes 16-31 K2,K3.
//   C/D 16x16:  VGPR r = row M (lanes 0-15) / M+8 (lanes 16-31), lane&15 = N.
// ---------------------------------------------------------------------
__global__ void fem_solid_assemble(const float* __restrict__ nodes,
                                   const int*   __restrict__ sel,
                                   const float* __restrict__ pE,
                                   const float* __restrict__ pNu,
                                   const float* __restrict__ pRho,
                                   float*       __restrict__ A) {
    __shared__ float sBt[8][16][8];   // B^T padded to 16x8 per wave
    __shared__ float sDB[8][8][16];   // D*B padded to 8x16 per wave
    __shared__ int   sDof[8][16];     // global dof ids (12 valid)

    const int w    = threadIdx.x >> 5;
    const int lane = threadIdx.x & 31;
    const int li   = lane & 15;
    const int half = lane >> 4;
    const int e    = blockIdx.x * 8 + w;   // 1250 blocks * 8 waves == 10000 exact

    const float E   = pE[0];
    const float nu  = pNu[0];
    const float rho = pRho[0];

    const int sl0 = sel[e * 4 + 0];
    const int sl1 = sel[e * 4 + 1];
    const int sl2 = sel[e * 4 + 2];
    const int sl3 = sel[e * 4 + 3];
    const int gbase = N_NODES_C - N_SOLID_C;

    const float* p0 = nodes + (size_t)(gbase + sl0) * 3;
    const float* p1 = nodes + (size_t)(gbase + sl1) * 3;
    const float* p2 = nodes + (size_t)(gbase + sl2) * 3;
    const float* p3 = nodes + (size_t)(gbase + sl3) * 3;
    float x0 = p0[0], y0 = p0[1], z0 = p0[2];
    float e1x = p1[0]-x0, e1y = p1[1]-y0, e1z = p1[2]-z0;
    float e2x = p2[0]-x0, e2y = p2[1]-y0, e2z = p2[2]-z0;
    float e3x = p3[0]-x0, e3y = p3[1]-y0, e3z = p3[2]-z0;

    float c1x = e2y*e3z - e2z*e3y, c1y = e2z*e3x - e2x*e3z, c1z = e2x*e3y - e2y*e3x;
    float det = e1x*c1x + e1y*c1y + e1z*c1z;
    float inv = 1.0f / det;
    float c2x = e3y*e1z - e3z*e1y, c2y = e3z*e1x - e3x*e1z, c2z = e3x*e1y - e3y*e1x;
    float c3x = e1y*e2z - e1z*e2y, c3y = e1z*e2x - e1x*e2z, c3z = e1x*e2y - e1y*e2x;

    float g1x = c1x*inv, g1y = c1y*inv, g1z = c1z*inv;
    float g2x = c2x*inv, g2y = c2y*inv, g2z = c2z*inv;
    float g3x = c3x*inv, g3y = c3y*inv, g3z = c3z*inv;
    float g0x = -(g1x+g2x+g3x), g0y = -(g1y+g2y+g3y), g0z = -(g1z+g2z+g3z);
    float Vs  = fabsf(det) * (1.0f / 6.0f);

    if (lane < 16) {
        int an = li / 3;              // node 0..3 (garbage for li>=12, masked)
        int d  = li - an * 3;         // dim 0..2
        float gax = (an == 0) ? g0x : (an == 1) ? g1x : (an == 2) ? g2x : g3x;
        float gay = (an == 0) ? g0y : (an == 1) ? g1y : (an == 2) ? g2y : g3y;
        float gaz = (an == 0) ? g0z : (an == 1) ? g1z : (an == 2) ? g2z : g3z;

        float b0 = 0.f, b1 = 0.f, b2 = 0.f, b3 = 0.f, b4 = 0.f, b5 = 0.f;
        if (li < 12) {
            if (d == 0)      { b0 = gax; b3 = gay; b5 = gaz; }
            else if (d == 1) { b1 = gay; b3 = gax; b4 = gaz; }
            else             { b2 = gaz; b4 = gay; b5 = gax; }
        }
        sBt[w][li][0] = b0; sBt[w][li][1] = b1; sBt[w][li][2] = b2;
        sBt[w][li][3] = b3; sBt[w][li][4] = b4; sBt[w][li][5] = b5;
        sBt[w][li][6] = 0.f; sBt[w][li][7] = 0.f;

        // isotropic D (Voigt): D*B column li
        float coeff = E / ((1.0f + nu) * (1.0f - 2.0f * nu));
        float d1 = coeff * (1.0f - nu);
        float d2 = coeff * nu;
        float d3 = coeff * 0.5f * (1.0f - 2.0f * nu);
        sDB[w][0][li] = d1 * b0 + d2 * (b1 + b2);
        sDB[w][1][li] = d1 * b1 + d2 * (b0 + b2);
        sDB[w][2][li] = d1 * b2 + d2 * (b0 + b1);
        sDB[w][3][li] = d3 * b3;
        sDB[w][4][li] = d3 * b4;
        sDB[w][5][li] = d3 * b5;
        sDB[w][6][li] = 0.f;
        sDB[w][7][li] = 0.f;

        int node = (an == 0) ? sl0 : (an == 1) ? sl1 : (an == 2) ? sl2 : sl3;
        sDof[w][li] = (li < 12) ? (NB_C + node * 3 + d) : 0;
    }
    __syncthreads();

    // Fragment loads per CDNA5 16x16x4 f32 layouts.
    v2f a0, a1, b0v, b1v;
    a0.x  = sBt[w][li][half * 2 + 0];
    a0.y  = sBt[w][li][half * 2 + 1];
    a1.x  = sBt[w][li][4 + half * 2 + 0];
    a1.y  = sBt[w][li][4 + half * 2 + 1];
    b0v.x = sDB[w][half * 2 + 0][li];
    b0v.y = sDB[w][half * 2 + 1][li];
    b1v.x = sDB[w][4 + half * 2 + 0][li];
    b1v.y = sDB[w][4 + half * 2 + 1][li];

    v8f acc = {0.f, 0.f, 0.f, 0.f, 0.f, 0.f, 0.f, 0.f};
    // D(16x16) = Bt(16x4) * DB(4x16) + C, chained over two K-chunks.
    acc = __builtin_amdgcn_wmma_f32_16x16x4_f32(false, a0, false, b0v,
                                                (short)0, acc, false, false);
    acc = __builtin_amdgcn_wmma_f32_16x16x4_f32(false, a1, false, b1v,
                                                (short)0, acc, false, false);

    float massd = OM2F * rho * Vs * 0.25f;   // omega^2 * (rho*Vs/4) lumped mass
    int   col   = sDof[w][li];
    bool  colOK = (li < 12);

#pragma unroll
    for (int r = 0; r < 8; ++r) {
        int  M   = half ? (r + 8) : r;
        bool mOK = half ? (r < 4) : true;
        if (colOK && mOK) {
            int   row = sDof[w][M];
            float v   = Vs * acc[r];
            if (M == li) v -= massd;       // element-local diagonal
            atomicAdd(&A[(size_t)row * NT_C + col], v);
        }
    }
}

// ---------------------------------------------------------------------
// Kernel 6: solid coupling penalty. Faithful to the reference quirk:
// always uses interface_normals[0] for every interface node.
// ---------------------------------------------------------------------
__global__ void fem_solid_penalty(const int*   __restrict__ isl,
                                  const float* __restrict__ normals,
                                  float*       __restrict__ A) {
    int t = blockIdx.x * blockDim.x + threadIdx.x;
    if (t >= N_IFACE_C * 9) return;
    int i = t / 9;
    int r = (t % 9) / 3;
    int c = t % 3;
    float val = PENALTY_F * normals[r] * normals[c];  // outer(n0, n0)
    int nd  = isl[i];
    int row = NB_C + nd * 3 + r;
    int col = NB_C + nd * 3 + c;
    atomicAdd(&A[(size_t)row * NT_C + col], val);
}

// ---------------------------------------------------------------------
extern "C" void kernel_launch(void* const* d_in, const int* in_sizes, int n_in,
                              void* d_out, int out_size, void* d_ws, size_t ws_size,
                              hipStream_t stream) {
    const float* nodes   = (const float*)d_in[0];
    const int*   fel     = (const int*)  d_in[1];
    const int*   sel     = (const int*)  d_in[2];
    const int*   imap    = (const int*)  d_in[3];
    const int*   isl     = (const int*)  d_in[4];
    const float* inorm   = (const float*)d_in[5];
    const int*   nearidx = (const int*)  d_in[6];
    const float* pE      = (const float*)d_in[7];
    const float* pNu     = (const float*)d_in[8];
    const float* pRho    = (const float*)d_in[9];

    float* A = (float*)d_out;
    float* F = A + (size_t)NT_C * NT_C;

    // 1) zero-fill A_g + F_g (784 MB: the bandwidth floor of this problem)
    long long n4 = ((long long)NT_C * NT_C + NT_C) / 4;   // divisible by 4
    fem_zero_out<<<2048, 256, 0, stream>>>((float4*)d_out, n4);

    // 2) fluid element scatter-add
    fem_fluid_assemble<<<(NF_ELEM_C + 255) / 256, 256, 0, stream>>>(nodes, fel, A);

    // 3) Dirichlet rows (must follow fluid assembly)
    dim3 dgrid((N_NODES_C + 255) / 256, N_NEAR_C);
    fem_dirichlet<<<dgrid, 256, 0, stream>>>(nearidx, A, F);

    // 4) fluid-side coupling penalty (must follow Dirichlet)
    fem_fluid_penalty<<<(N_IFACE_C + 255) / 256, 256, 0, stream>>>(imap, A);

    // 5) solid element assembly via WMMA (one wave32 per tet; 1250*8 == 10000)
    fem_solid_assemble<<<NS_ELEM_C / 8, 256, 0, stream>>>(nodes, sel, pE, pNu, pRho, A);

    // 6) solid-side coupling penalty
    fem_solid_penalty<<<(N_IFACE_C * 9 + 255) / 256, 256, 0, stream>>>(isl, inorm, A);
}